// SummarizationTrace_22058952032948
// MI455X (gfx1250) — compile-verified
//
#include <hip/hip_runtime.h>

typedef __attribute__((ext_vector_type(2))) float v2f;
typedef __attribute__((ext_vector_type(4))) float v4f;
typedef __attribute__((ext_vector_type(8))) float v8f;

#define PROG  256
#define GRIDD 512
#define NTOK  4096          // B*T
#define E_TOT 65536         // B*T*16
#define OUTD  768           // PROG + GRIDD
#define CHUNK 256

__device__ __forceinline__ int lower_bound_i32(const int* __restrict__ a, int n, int key) {
    int lo = 0, hi = n;
    while (lo < hi) {
        int mid = (lo + hi) >> 1;
        if (a[mid] < key) lo = mid + 1; else hi = mid;
    }
    return lo;
}

__global__ __launch_bounds__(256)
void summ_trace_wmma(const float* __restrict__ inp,
                     const float* __restrict__ trace,
                     const float* __restrict__ weights,
                     const int*   __restrict__ seg_ids,
                     const int*   __restrict__ trace_rows,
                     const int*   __restrict__ time_idx,
                     float*       __restrict__ out) {
    // per-entry metadata record: .x = weight bits, .y = seg id, .z = row element offset
    __shared__ int4 s_meta[CHUNK];

    const int tid     = threadIdx.x;
    const int wave    = tid >> 5;
    const int lane    = tid & 31;
    const int khalf2  = (lane >> 4) * 2;  // 0 -> K slots {0,1}; 2 -> K slots {2,3}
    const int mrow    = lane & 15;        // A-matrix row (and B-matrix column index)
    const int tokBase = blockIdx.x * 16;
    const int colBase = blockIdx.y * 256; // column split across gridDim.y == 2
    const int col0    = colBase + wave * 32;  // this wave: 2 tiles, cols [col0, col0+32)
    const int colp    = col0 + mrow;
    const int tokm    = tokBase + mrow;

    // ---- concat part 1: out[tok, 0:256] = inp_embed[tok, :] (only the y==0 slice)
    if (blockIdx.y == 0) {
        const v4f* src = (const v4f*)(inp + (size_t)tokBase * PROG);
        #pragma unroll
        for (int r = 0; r < 4; ++r) {
            int i4  = tid + r * 256;      // 0..1023; 64 float4 per token
            int tok = i4 >> 6;
            int c4  = i4 & 63;
            v4f val = src[i4];
            *(v4f*)(out + (size_t)(tokBase + tok) * OUTD + c4 * 4) = val;
        }
    }

    // ---- ragged entry range for this 16-token tile (seg_ids is sorted)
    const int lo    = lower_bound_i32(seg_ids, E_TOT, tokBase);
    const int hi    = lower_bound_i32(seg_ids, E_TOT, tokBase + 16);
    const int total = hi - lo;

    v8f acc0 = {}, acc1 = {};

    for (int cs = 0; cs < total; cs += CHUNK) {
        int nk = total - cs; if (nk > CHUNK) nk = CHUNK;
        __syncthreads();                  // protect LDS reuse across chunks
        // Stage ALL 256 slots: tails get a neutral record (seg=-1 => A column all zero,
        // w=0, off=0 => B loads a valid-but-ignored row). Hot loop is branch-free and
        // we can round the K trip count up to a multiple of 8 with no tail code.
        int4 m = make_int4(0, -1, 0, 0);
        if (tid < nk) {
            int e = lo + cs + tid;
            m.x = __float_as_int(weights[e]);
            m.y = seg_ids[e];
            m.z = (trace_rows[e] * 128 + time_idx[e]) << 9;   // * GRIDD
        }
        s_meta[tid] = m;
        __syncthreads();

        const int nkR = (nk + 7) & ~7;    // multiple of 8, <= 256; pad slots are neutral
        for (int k0 = 0; k0 < nkR; k0 += 8) {
            #pragma unroll
            for (int u = 0; u < 2; ++u) {
                const int ks = k0 + u * 4;
                int4 m0 = s_meta[ks + khalf2];
                int4 m1 = s_meta[ks + khalf2 + 1];

                // sel in {0.0f,1.0f}; the multiply forces an unconditional weight load
                // (w*0 is not foldable to a select without fast-math), avoiding the
                // EXEC-masked divergent ds_load the ternary form produced.
                float sel0 = (m0.y == tokm) ? 1.0f : 0.0f;
                float sel1 = (m1.y == tokm) ? 1.0f : 0.0f;
                v2f a;
                a.x = __int_as_float(m0.x) * sel0;
                a.y = __int_as_float(m1.x) * sel1;

                const float* r0 = trace + m0.z;
                const float* r1 = trace + m1.z;

                v2f b0, b1;
                b0.x = r0[colp];        b0.y = r1[colp];
                b1.x = r0[colp + 16];   b1.y = r1[colp + 16];

                acc0 = __builtin_amdgcn_wmma_f32_16x16x4_f32(false, a, false, b0,
                                                             (short)0, acc0, false, false);
                acc1 = __builtin_amdgcn_wmma_f32_16x16x4_f32(false, a, false, b1,
                                                             (short)0, acc1, false, false);
            }
        }
    }

    // ---- concat part 2: out[tok, 256:768] (C/D layout: row M = v + khalf*8, col = lane&15)
    float* obase = out + (size_t)tokBase * OUTD + PROG;
    #pragma unroll
    for (int v = 0; v < 8; ++v) {
        int M = v + (khalf2 >> 1) * 8;
        float* orow = obase + (size_t)M * OUTD;
        orow[colp]      = acc0[v];
        orow[colp + 16] = acc1[v];
    }
}

extern "C" void kernel_launch(void* const* d_in, const int* in_sizes, int n_in,
                              void* d_out, int out_size, void* d_ws, size_t ws_size,
                              hipStream_t stream) {
    const float* inp        = (const float*)d_in[0];
    const float* trace      = (const float*)d_in[1];
    const float* weights    = (const float*)d_in[2];
    const int*   seg_ids    = (const int*)d_in[3];
    const int*   trace_rows = (const int*)d_in[4];
    const int*   time_idx   = (const int*)d_in[5];
    float*       out        = (float*)d_out;

    dim3 grid(NTOK / 16, 2);  // 256 token tiles x 2 column halves
    dim3 block(256);          // 8 wave32s; each wave owns 32 of this half's 256 columns
    summ_trace_wmma<<<grid, block, 0, stream>>>(inp, trace, weights,
                                                seg_ids, trace_rows, time_idx, out);
}